// MultiQueryAttention_71244917506184
// MI455X (gfx1250) — compile-verified
//
#include <hip/hip_runtime.h>

typedef __attribute__((ext_vector_type(2))) float v2f;
typedef __attribute__((ext_vector_type(8))) float v8f;

#define EMB   1024
#define HEADS 16
#define HD    64
#define SEQ   2048

static __device__ __forceinline__ v8f wmma_f32x4(v2f a, v2f b, v8f c) {
  // D = A(16x4) * B(4x16) + C, all fp32
  return __builtin_amdgcn_wmma_f32_16x16x4_f32(false, a, false, b, (short)0, c,
                                               false, false);
}

// C[M,N] = X[M,K] @ W[K,N].  One wave computes a 16x64 output strip
// (4 accumulators share one A fragment).  Block = 8 waves = 128 rows.
__global__ void __launch_bounds__(256) gemm16x64(const float* __restrict__ X,
                                                 const float* __restrict__ W,
                                                 float* __restrict__ Out,
                                                 int N, int Kd) {
  const int wave = threadIdx.x >> 5;
  const int lane = threadIdx.x & 31;
  const int half = lane >> 4;   // which K-pair / M-half
  const int l16  = lane & 15;
  const int rowTile = blockIdx.y * 128 + wave * 16;
  const int col0    = blockIdx.x * 64;

  v8f acc[4] = {};
  const float* xr = X + (size_t)(rowTile + l16) * Kd;
  for (int k = 0; k < Kd; k += 4) {
    const int k0 = k + half * 2;
    v2f a;
    a.x = xr[k0];
    a.y = xr[k0 + 1];
#pragma unroll
    for (int nc = 0; nc < 4; ++nc) {
      const int col = col0 + nc * 16 + l16;
      v2f b;
      b.x = W[(size_t)k0 * N + col];
      b.y = W[(size_t)(k0 + 1) * N + col];
      acc[nc] = wmma_f32x4(a, b, acc[nc]);
    }
  }
#pragma unroll
  for (int nc = 0; nc < 4; ++nc)
#pragma unroll
    for (int r = 0; r < 8; ++r)
      Out[(size_t)(rowTile + r + 8 * half) * N + col0 + nc * 16 + l16] =
          acc[nc][r];
}

// Causal flash attention, one wave handles a 16-row Q tile of one (b,h).
// Q: [B,S,EMB] (head h = cols h*64..h*64+63), K/V: [B,S,64], Out: [B,S,EMB].
__global__ void __launch_bounds__(32) mqa_flash_attn(const float* __restrict__ Q,
                                                     const float* __restrict__ K,
                                                     const float* __restrict__ V,
                                                     float* __restrict__ Out) {
  const int it   = blockIdx.x;  // query tile index (S/16)
  const int h    = blockIdx.y;
  const int b    = blockIdx.z;
  const int lane = threadIdx.x & 31;
  const int half = lane >> 4;
  const int l16  = lane & 15;

  __shared__ float pbuf[16 * 16];  // P tile staging for C->A layout transpose

  const float* Kb = K + (size_t)b * SEQ * HD;
  const float* Vb = V + (size_t)b * SEQ * HD;

  // Q fragments (A layout), pre-scaled by 1/sqrt(64) = 0.125
  v2f qf[16];
  const float* qrow = Q + ((size_t)b * SEQ + it * 16 + l16) * EMB + h * HD;
#pragma unroll
  for (int kk = 0; kk < 16; ++kk) {
    const int d0 = 4 * kk + half * 2;
    qf[kk].x = qrow[d0] * 0.125f;
    qf[kk].y = qrow[d0 + 1] * 0.125f;
  }

  float m_i[8], l_i[8];
  v8f o[4] = {};
#pragma unroll
  for (int r = 0; r < 8; ++r) {
    m_i[r] = -__builtin_inff();
    l_i[r] = 0.0f;
  }

  for (int j = 0; j <= it; ++j) {
    // ---- scores S = (Q*0.125) @ K_tile^T : 16x16, K-dim = 64 ----
    v8f s = {};
    const float* krow = Kb + (size_t)(j * 16 + l16) * HD;
#pragma unroll
    for (int kk = 0; kk < 16; ++kk) {
      const int d0 = 4 * kk + half * 2;
      v2f bf;
      bf.x = krow[d0];
      bf.y = krow[d0 + 1];
      s = wmma_f32x4(qf[kk], bf, s);
    }
    // causal mask (only the diagonal tile needs it)
    if (j == it) {
#pragma unroll
      for (int r = 0; r < 8; ++r)
        if (l16 > r + 8 * half) s[r] = -__builtin_inff();
    }
    // ---- online softmax (row stats across 16-lane groups) ----
    float p[8];
#pragma unroll
    for (int r = 0; r < 8; ++r) {
      float v = s[r];
      for (int off = 8; off >= 1; off >>= 1)
        v = fmaxf(v, __shfl_xor(v, off, 16));
      const float mnew = fmaxf(m_i[r], v);
      p[r] = __expf(s[r] - mnew);
      const float scale = __expf(m_i[r] - mnew);
      float rs = p[r];
      for (int off = 8; off >= 1; off >>= 1) rs += __shfl_xor(rs, off, 16);
      l_i[r] = l_i[r] * scale + rs;
      m_i[r] = mnew;
#pragma unroll
      for (int nc = 0; nc < 4; ++nc) o[nc][r] *= scale;
    }
    // ---- transpose P (C layout -> A layout) via LDS ----
#pragma unroll
    for (int r = 0; r < 8; ++r) pbuf[(r + 8 * half) * 16 + l16] = p[r];
    __syncthreads();
    // ---- O += P @ V_tile : 16x64, K-dim = 16 ----
#pragma unroll
    for (int st = 0; st < 4; ++st) {
      const int kk0 = 4 * st + half * 2;
      v2f pa;
      pa.x = pbuf[l16 * 16 + kk0];
      pa.y = pbuf[l16 * 16 + kk0 + 1];
#pragma unroll
      for (int nc = 0; nc < 4; ++nc) {
        v2f vb;
        vb.x = Vb[(size_t)(j * 16 + kk0) * HD + nc * 16 + l16];
        vb.y = Vb[(size_t)(j * 16 + kk0 + 1) * HD + nc * 16 + l16];
        o[nc] = wmma_f32x4(pa, vb, o[nc]);
      }
    }
    __syncthreads();
  }

  // ---- normalize and write out[b, s, h*64 + d] ----
#pragma unroll
  for (int r = 0; r < 8; ++r) {
    const float inv = 1.0f / l_i[r];
    const size_t row = (size_t)b * SEQ + it * 16 + r + 8 * half;
#pragma unroll
    for (int nc = 0; nc < 4; ++nc)
      Out[row * EMB + h * HD + nc * 16 + l16] = o[nc][r] * inv;
  }
}

extern "C" void kernel_launch(void* const* d_in, const int* in_sizes, int n_in,
                              void* d_out, int out_size, void* d_ws,
                              size_t ws_size, hipStream_t stream) {
  (void)n_in;
  (void)out_size;
  (void)ws_size;
  const float* X  = (const float*)d_in[0];  // [B,S,EMB]
  const float* Wq = (const float*)d_in[1];  // [EMB,EMB]
  const float* Wk = (const float*)d_in[2];  // [EMB,HD]
  const float* Wv = (const float*)d_in[3];  // [EMB,HD]
  float* Out = (float*)d_out;               // [B,S,EMB]

  const int BS = in_sizes[0] / EMB;  // B*S rows
  const int B  = BS / SEQ;

  float* qbuf = (float*)d_ws;                    // [BS, EMB]  16 MB
  float* kbuf = qbuf + (size_t)BS * EMB;         // [BS, HD]    1 MB
  float* vbuf = kbuf + (size_t)BS * HD;          // [BS, HD]    1 MB

  dim3 blk(256);
  dim3 gq(EMB / 64, BS / 128);
  gemm16x64<<<gq, blk, 0, stream>>>(X, Wq, qbuf, EMB, EMB);
  dim3 gkv(HD / 64, BS / 128);
  gemm16x64<<<gkv, blk, 0, stream>>>(X, Wk, kbuf, HD, EMB);
  gemm16x64<<<gkv, blk, 0, stream>>>(X, Wv, vbuf, HD, EMB);

  dim3 ga(SEQ / 16, HEADS, B);
  mqa_flash_attn<<<ga, dim3(32), 0, stream>>>(qbuf, kbuf, vbuf, Out);
}